// EdgeNetwork_70265664963042
// MI455X (gfx1250) — compile-verified
//
#include <hip/hip_runtime.h>
#include <cmath>

typedef __attribute__((ext_vector_type(2))) float v2f;
typedef __attribute__((ext_vector_type(8))) float v8f;

#define B_    8
#define N_    256
#define FIN   64
#define MID_  96
#define FOUT  64
#define DIN   129      // 2*FIN + 1
#define DINP  132      // padded to multiple of 4 (pad rows are zero)
#define HPAD  100      // padded row stride for h tiles (keeps 8B alignment)
#define W1S   224      // pair-row stride for sW1 (== 32 mod 64 dwords: conflict-free)
#define W2S   160      // pair-row stride for sW2 (== 32 mod 64 dwords)

// One workgroup per (b, n): builds the 255x129 edge-block implicitly,
// GEMM1 (129->96, relu) and GEMM2 (96->64, tanh) via fp32 WMMA, then
// reduces the 255 rows into out[b, n, 0:64].
// Weights live in LDS K-pair-interleaved: {W[2p][c], W[2p+1][c]} contiguous,
// so each WMMA B-fragment is a single aligned ds_load_b64 (no repack movs).
__global__ __launch_bounds__(256) void edge_net_kernel(
    const float* __restrict__ node,   // [B, N, FIN]
    const float* __restrict__ edge,   // [B, N, N-1]
    const float* __restrict__ W1,     // [DIN, MID]
    const float* __restrict__ b1,     // [MID]
    const float* __restrict__ W2,     // [MID, FOUT]
    const float* __restrict__ b2,     // [FOUT]
    float* __restrict__ out)          // [B, N, FOUT]
{
    __shared__ float sW1[(DINP / 2) * W1S];   // 66*224*4 = 59136 B
    __shared__ float sW2[(MID_ / 2) * W2S];   // 48*160*4 = 30720 B
    __shared__ float sH[8 * 16 * HPAD];       // 51200 B (per-wave 16x96 h tile)
    __shared__ float sOut[FOUT];              // 256 B

    const int tid   = threadIdx.x;
    const int wave  = tid >> 5;
    const int lane  = tid & 31;
    const int lhalf = lane >> 4;            // 0 or 1 (selects K pair / row half)
    const int l16   = lane & 15;

    const int blk = blockIdx.x;
    const int b   = blk >> 8;
    const int n   = blk & 255;

    // ---- stage weights into LDS, K-pair interleaved ----
    for (int i = tid; i < DINP * MID_; i += 256) {
        const int k = i / MID_, c = i % MID_;
        sW1[(k >> 1) * W1S + c * 2 + (k & 1)] = (k < DIN) ? W1[i] : 0.0f;
    }
    for (int i = tid; i < MID_ * FOUT; i += 256) {
        const int k = i / FOUT, c = i % FOUT;
        sW2[(k >> 1) * W2S + c * 2 + (k & 1)] = W2[i];
    }
    if (tid < FOUT) sOut[tid] = 0.0f;
    __syncthreads();

    // per-lane bias preloads (uniform per output column)
    float bias1[6];
#pragma unroll
    for (int t = 0; t < 6; ++t) bias1[t] = b1[t * 16 + l16];
    float bias2[4];
#pragma unroll
    for (int t = 0; t < 4; ++t) bias2[t] = b2[t * 16 + l16];

    const long  nodeB = (long)b * N_ * FIN;
    const float* nodeN = node + nodeB + (long)n * FIN;
    float* myH = sH + wave * (16 * HPAD);

    // Opaque (to LICM) base offsets for the LDS weight reads: prevents the
    // compiler from promoting all 294 B-fragments to registers across the
    // tile loop (432-VGPR allocation / scratch spills otherwise).
    int w1base = 0;
    int w2base = 0;

    // 16 row tiles of 16 (255 rows padded to 256); 8 waves, 2 tiles each
    for (int iter = 0; iter < 2; ++iter) {
        asm volatile("" : "+s"(w1base), "+s"(w2base));

        const int tile = wave + iter * 8;
        const int mrow = tile * 16 + l16;                         // A-matrix row this lane feeds
        const int jm   = min(mrow + (mrow >= n ? 1 : 0), N_ - 1); // neighbor (clamped for pad row)
        const float edgev = edge[((long)b * N_ + n) * (N_ - 1) + min(mrow, N_ - 2)];
        const float* nodeJ = node + nodeB + (long)jm * FIN;

        // ---- GEMM1: [16 x 129] @ [129 x 96] ----
        v8f acc1[6];
#pragma unroll
        for (int t = 0; t < 6; ++t) acc1[t] = (v8f){};

        // segment 1: k in [0, 64) -- A rows are the (uniform) node_n features
#pragma unroll
        for (int k0 = 0; k0 < 64; k0 += 4) {
            const int p = (k0 >> 1) + lhalf;                      // K-pair index
            const v2f a = *(const v2f*)(nodeN + k0 + 2 * lhalf);  // global_load_b64
#pragma unroll
            for (int t = 0; t < 6; ++t) {
                const int col = t * 16 + l16;
                const v2f bf = *(const v2f*)(sW1 + w1base + p * W1S + col * 2); // ds_load_b64
                acc1[t] = __builtin_amdgcn_wmma_f32_16x16x4_f32(
                    false, a, false, bf, (short)0, acc1[t], false, false);
            }
        }
        // segment 2: k in [64, 128) -- A rows are per-row neighbor features
#pragma unroll
        for (int k0 = 0; k0 < 64; k0 += 4) {
            const int p = 32 + (k0 >> 1) + lhalf;
            const v2f a = *(const v2f*)(nodeJ + k0 + 2 * lhalf);  // global_load_b64
#pragma unroll
            for (int t = 0; t < 6; ++t) {
                const int col = t * 16 + l16;
                const v2f bf = *(const v2f*)(sW1 + w1base + p * W1S + col * 2);
                acc1[t] = __builtin_amdgcn_wmma_f32_16x16x4_f32(
                    false, a, false, bf, (short)0, acc1[t], false, false);
            }
        }
        // tail: k = 128 (edge scalar; k = 129..131 are zero pad)
        {
            const int p = 64 + lhalf;
            v2f a;
            a.x = (lhalf == 0) ? edgev : 0.0f;                    // v_cndmask, no branch
            a.y = 0.0f;
#pragma unroll
            for (int t = 0; t < 6; ++t) {
                const int col = t * 16 + l16;
                const v2f bf = *(const v2f*)(sW1 + w1base + p * W1S + col * 2);
                acc1[t] = __builtin_amdgcn_wmma_f32_16x16x4_f32(
                    false, a, false, bf, (short)0, acc1[t], false, false);
            }
        }

        // bias + relu; spill h tile to per-wave LDS (C-layout -> row-major)
#pragma unroll
        for (int t = 0; t < 6; ++t) {
            const int col = t * 16 + l16;
#pragma unroll
            for (int v = 0; v < 8; ++v) {
                const int M = v + 8 * lhalf;
                myH[M * HPAD + col] = fmaxf(acc1[t][v] + bias1[t], 0.0f);
            }
        }

        // ---- GEMM2: [16 x 96] @ [96 x 64] ----
        v8f acc2[4];
#pragma unroll
        for (int t = 0; t < 4; ++t) acc2[t] = (v8f){};

#pragma unroll
        for (int k0 = 0; k0 < 96; k0 += 4) {
            const int p  = (k0 >> 1) + lhalf;
            const int kb = k0 + 2 * lhalf;
            const v2f a = *(const v2f*)(myH + l16 * HPAD + kb);   // ds_load_b64
#pragma unroll
            for (int t = 0; t < 4; ++t) {
                const int col = t * 16 + l16;
                const v2f bf = *(const v2f*)(sW2 + w2base + p * W2S + col * 2); // ds_load_b64
                acc2[t] = __builtin_amdgcn_wmma_f32_16x16x4_f32(
                    false, a, false, bf, (short)0, acc2[t], false, false);
            }
        }

        // tanh(+bias2), mask pad row (m == 255), reduce rows into sOut
#pragma unroll
        for (int t = 0; t < 4; ++t) {
            float partial = 0.0f;
#pragma unroll
            for (int v = 0; v < 8; ++v) {
                const int rowm = tile * 16 + v + 8 * lhalf;
                const float val = tanhf(acc2[t][v] + bias2[t]);
                partial += (rowm < N_ - 1) ? val : 0.0f;
            }
            atomicAdd(&sOut[t * 16 + l16], partial);              // ds_add_f32
        }
    }

    __syncthreads();
    if (tid < FOUT)
        out[((long)b * N_ + n) * FOUT + tid] = sOut[tid];
}

extern "C" void kernel_launch(void* const* d_in, const int* in_sizes, int n_in,
                              void* d_out, int out_size, void* d_ws, size_t ws_size,
                              hipStream_t stream) {
    const float* node  = (const float*)d_in[0];  // inp_node  [8,256,64]
    const float* edgep = (const float*)d_in[1];  // inp_edge  [8,256,255]
    const float* W1    = (const float*)d_in[2];  // [129,96]
    const float* b1    = (const float*)d_in[3];  // [96]
    const float* W2    = (const float*)d_in[4];  // [96,64]
    const float* b2    = (const float*)d_in[5];  // [64]
    float* out = (float*)d_out;                  // [8,256,64]

    dim3 grid(B_ * N_);   // 2048 workgroups, one per (b, n)
    dim3 block(256);      // 8 wave32 per WG
    hipLaunchKernelGGL(edge_net_kernel, grid, block, 0, stream,
                       node, edgep, W1, b1, W2, b2, out);
}